// GraphEncoder_23398981828829
// MI455X (gfx1250) — compile-verified
//
#include <hip/hip_runtime.h>
#include <math.h>

// ---------------------------------------------------------------------------
// GCN (3x GCNConv + global mean pool) for MI455X / gfx1250.
// Fixed problem sizes from the reference:
#define N_NODES  50000
#define N_EDGES  800000
#define IN_DIM   128
#define HID      64
#define OUT_DIM  128
#define N_GRAPHS 256
// ---------------------------------------------------------------------------

typedef float v2f __attribute__((ext_vector_type(2)));
typedef float v8f __attribute__((ext_vector_type(8)));

// ------------------------------ utility ------------------------------------

__global__ void fill_f32(float* __restrict__ p, float v, long n) {
    long i = (long)blockIdx.x * blockDim.x + threadIdx.x;
    if (i < n) p[i] = v;
}

// degree with self-loop: deg initialized to 1.0, +1 per incoming edge
__global__ void deg_kernel(const long long* __restrict__ dst64,
                           float* __restrict__ deg, int E) {
    int e = blockIdx.x * blockDim.x + threadIdx.x;
    if (e < E) atomicAdd(&deg[(int)dst64[e]], 1.0f);
}

// dinv = rsqrt(deg)   (deg >= 1 always due to self-loops)
__global__ void dinv_kernel(float* __restrict__ deg, int N) {
    int n = blockIdx.x * blockDim.x + threadIdx.x;
    if (n < N) deg[n] = rsqrtf(deg[n]);
}

// int64 -> int32 edge index + per-edge symmetric norm
__global__ void edge_prep(const long long* __restrict__ src64,
                          const long long* __restrict__ dst64,
                          const float* __restrict__ dinv,
                          int* __restrict__ src32, int* __restrict__ dst32,
                          float* __restrict__ norm, int E) {
    int e = blockIdx.x * blockDim.x + threadIdx.x;
    if (e >= E) return;
    int s = (int)src64[e];
    int d = (int)dst64[e];
    src32[e] = s;
    dst32[e] = d;
    norm[e] = dinv[s] * dinv[d];
}

// ------------------------------ WMMA GEMM ----------------------------------
// H[N,F] = X[N,K] @ W[K,F], all row-major f32.
// One wave32 per 16x16 output tile, V_WMMA_F32_16X16X4_F32 along K.
// Requires: N%16==0, K%4==0, F%16==0 (true for all three layers).

__global__ void gemm_wmma_f32(const float* __restrict__ X,
                              const float* __restrict__ W,
                              float* __restrict__ H,
                              int Nn, int K, int F) {
    int wave = (int)((blockIdx.x * blockDim.x + threadIdx.x) >> 5);
    int lane = threadIdx.x & 31;
    int tilesN = F >> 4;
    int totalTiles = (Nn >> 4) * tilesN;
    if (wave >= totalTiles) return;          // wave-uniform: EXEC stays full
    int mt = wave / tilesN;
    int nt = wave % tilesN;

    int half = lane >> 4;                    // 0: lanes 0-15, 1: lanes 16-31
    int l    = lane & 15;
    int row  = (mt << 4) + l;                // A-matrix row for this lane
    int col  = (nt << 4) + l;                // B/C/D column for this lane

    v8f acc = {};
    const float* xrow = X + (size_t)row * K;
    for (int k = 0; k < K; k += 4) {
        int ka = k + (half << 1);            // lanes 0-15 -> K{k,k+1}; 16-31 -> K{k+2,k+3}
        v2f a, b;
        a.x = xrow[ka];
        a.y = xrow[ka + 1];
        b.x = W[(size_t)ka * F + col];
        b.y = W[(size_t)(ka + 1) * F + col];
        acc = __builtin_amdgcn_wmma_f32_16x16x4_f32(
            /*neg_a=*/false, a, /*neg_b=*/false, b,
            /*c_mod=*/(short)0, acc, /*reuse_a=*/false, /*reuse_b=*/false);
    }

    // D layout: VGPR v, lanes 0-15 -> M=v; lanes 16-31 -> M=v+8; N=l
    int mbase = (mt << 4) + (half << 3);
#pragma unroll
    for (int v = 0; v < 8; ++v) {
        H[(size_t)(mbase + v) * F + col] = acc[v];
    }
}

// ------------------------------ edge scatter --------------------------------
// AGG[dst] += norm * H[src], 4 features per thread (L2-resident f32 atomics)

__global__ void scatter_edges(const int* __restrict__ src,
                              const int* __restrict__ dst,
                              const float* __restrict__ norm,
                              const float* __restrict__ H,
                              float* __restrict__ AGG, int E, int F) {
    int t = blockIdx.x * blockDim.x + threadIdx.x;
    int chunks = F >> 2;
    int e = t / chunks;
    if (e >= E) return;
    int f4 = (t - e * chunks) << 2;
    float nrm = norm[e];
    const float4 h4 = *(const float4*)(H + (size_t)src[e] * F + f4);
    float* a = AGG + (size_t)dst[e] * F + f4;
    atomicAdd(a + 0, nrm * h4.x);
    atomicAdd(a + 1, nrm * h4.y);
    atomicAdd(a + 2, nrm * h4.z);
    atomicAdd(a + 3, nrm * h4.w);
}

// ------------------------------ epilogue ------------------------------------
// out = [gelu]( AGG[n,f] + dinv[n]^2 * H[n,f] + b[f] )
// (self-loop contribution folded here deterministically — no atomics needed)

__device__ __forceinline__ float gelu_tanh(float x) {
    const float c = 0.7978845608028654f;   // sqrt(2/pi)
    float x3 = x * x * x;
    return 0.5f * x * (1.0f + tanhf(c * (x + 0.044715f * x3)));
}

template <int DO_GELU>
__global__ void epilogue(const float* __restrict__ AGG,
                         const float* __restrict__ H,
                         const float* __restrict__ dinv,
                         const float* __restrict__ bias,
                         float* __restrict__ OUT, int N, int F) {
    long idx = (long)blockIdx.x * blockDim.x + threadIdx.x;
    long total = (long)N * F;
    if (idx >= total) return;
    int n = (int)(idx / F);
    int f = (int)(idx - (long)n * F);
    float di = dinv[n];
    float v = AGG[idx] + di * di * H[idx] + bias[f];
    OUT[idx] = DO_GELU ? gelu_tanh(v) : v;
}

// ------------------------------ pooling -------------------------------------

__global__ void pool_accum(const long long* __restrict__ batch,
                           const float* __restrict__ h,
                           float* __restrict__ sums,
                           float* __restrict__ counts, int N, int F) {
    int t = blockIdx.x * blockDim.x + threadIdx.x;
    int chunks = F >> 2;
    int n = t / chunks;
    if (n >= N) return;
    int f4 = (t - n * chunks) << 2;
    int g = (int)batch[n];
    const float4 v = *(const float4*)(h + (size_t)n * F + f4);
    float* s = sums + (size_t)g * F + f4;
    atomicAdd(s + 0, v.x);
    atomicAdd(s + 1, v.y);
    atomicAdd(s + 2, v.z);
    atomicAdd(s + 3, v.w);
    if (f4 == 0) atomicAdd(&counts[g], 1.0f);
}

__global__ void pool_div(const float* __restrict__ sums,
                         const float* __restrict__ counts,
                         float* __restrict__ out, int G, int F) {
    int idx = blockIdx.x * blockDim.x + threadIdx.x;
    if (idx >= G * F) return;
    int g = idx / F;
    out[idx] = sums[idx] / fmaxf(counts[g], 1.0f);
}

// ------------------------------ launcher ------------------------------------

static inline int cdiv(long a, int b) { return (int)((a + b - 1) / b); }

extern "C" void kernel_launch(void* const* d_in, const int* in_sizes, int n_in,
                              void* d_out, int out_size, void* d_ws, size_t ws_size,
                              hipStream_t stream) {
    (void)in_sizes; (void)n_in; (void)out_size; (void)ws_size;

    const float*     x      = (const float*)d_in[0];           // [N, 128]
    const long long* eidx   = (const long long*)d_in[1];       // [2, E] int64
    const long long* batch  = (const long long*)d_in[2];       // [N] int64
    // d_in[3] = num_graphs (scalar) — compile-time constant here
    const float* W1 = (const float*)d_in[4];                   // [128, 64]
    const float* b1 = (const float*)d_in[5];                   // [64]
    const float* W2 = (const float*)d_in[6];                   // [64, 64]
    const float* b2 = (const float*)d_in[7];                   // [64]
    const float* W3 = (const float*)d_in[8];                   // [64, 128]
    const float* b3 = (const float*)d_in[9];                   // [128]
    float* out = (float*)d_out;                                // [256, 128]

    const long long* src64 = eidx;
    const long long* dst64 = eidx + N_EDGES;

    // --- workspace carving (all chunks 16B-aligned) ---
    char* p = (char*)d_ws;
    float* dinv  = (float*)p;  p += (size_t)N_NODES * 4;             // deg -> dinv (in place)
    int*   src32 = (int*)p;    p += (size_t)N_EDGES * 4;
    int*   dst32 = (int*)p;    p += (size_t)N_EDGES * 4;
    float* norm  = (float*)p;  p += (size_t)N_EDGES * 4;
    float* Hbuf  = (float*)p;  p += (size_t)N_NODES * 128 * 4;
    float* AGG   = (float*)p;  p += (size_t)N_NODES * 128 * 4;
    float* ACT   = (float*)p;  p += (size_t)N_NODES * 128 * 4;
    float* sums  = (float*)p;  p += (size_t)N_GRAPHS * 128 * 4;      // sums then counts (contiguous)
    float* cnts  = (float*)p;  p += (size_t)N_GRAPHS * 4;

    const int B = 256;

    // --- GCN normalization (shared across all three layers) ---
    fill_f32<<<cdiv(N_NODES, B), B, 0, stream>>>(dinv, 1.0f, N_NODES); // self-loop deg
    deg_kernel<<<cdiv(N_EDGES, B), B, 0, stream>>>(dst64, dinv, N_EDGES);
    dinv_kernel<<<cdiv(N_NODES, B), B, 0, stream>>>(dinv, N_NODES);
    edge_prep<<<cdiv(N_EDGES, B), B, 0, stream>>>(src64, dst64, dinv,
                                                  src32, dst32, norm, N_EDGES);

    // --- layer 1: [N,128]@[128,64] -> scatter -> gelu ---
    {
        long waves = (long)(N_NODES / 16) * (HID / 16);
        gemm_wmma_f32<<<cdiv(waves * 32, B), B, 0, stream>>>(x, W1, Hbuf,
                                                             N_NODES, IN_DIM, HID);
        fill_f32<<<cdiv((long)N_NODES * HID, B), B, 0, stream>>>(AGG, 0.0f,
                                                                 (long)N_NODES * HID);
        scatter_edges<<<cdiv((long)N_EDGES * (HID / 4), B), B, 0, stream>>>(
            src32, dst32, norm, Hbuf, AGG, N_EDGES, HID);
        epilogue<1><<<cdiv((long)N_NODES * HID, B), B, 0, stream>>>(
            AGG, Hbuf, dinv, b1, ACT, N_NODES, HID);
    }

    // --- layer 2: [N,64]@[64,64] -> scatter -> gelu ---
    {
        long waves = (long)(N_NODES / 16) * (HID / 16);
        gemm_wmma_f32<<<cdiv(waves * 32, B), B, 0, stream>>>(ACT, W2, Hbuf,
                                                             N_NODES, HID, HID);
        fill_f32<<<cdiv((long)N_NODES * HID, B), B, 0, stream>>>(AGG, 0.0f,
                                                                 (long)N_NODES * HID);
        scatter_edges<<<cdiv((long)N_EDGES * (HID / 4), B), B, 0, stream>>>(
            src32, dst32, norm, Hbuf, AGG, N_EDGES, HID);
        epilogue<1><<<cdiv((long)N_NODES * HID, B), B, 0, stream>>>(
            AGG, Hbuf, dinv, b2, ACT, N_NODES, HID);
    }

    // --- layer 3: [N,64]@[64,128] -> scatter -> bias only ---
    {
        long waves = (long)(N_NODES / 16) * (OUT_DIM / 16);
        gemm_wmma_f32<<<cdiv(waves * 32, B), B, 0, stream>>>(ACT, W3, Hbuf,
                                                             N_NODES, HID, OUT_DIM);
        fill_f32<<<cdiv((long)N_NODES * OUT_DIM, B), B, 0, stream>>>(
            AGG, 0.0f, (long)N_NODES * OUT_DIM);
        scatter_edges<<<cdiv((long)N_EDGES * (OUT_DIM / 4), B), B, 0, stream>>>(
            src32, dst32, norm, Hbuf, AGG, N_EDGES, OUT_DIM);
        epilogue<0><<<cdiv((long)N_NODES * OUT_DIM, B), B, 0, stream>>>(
            AGG, Hbuf, dinv, b3, ACT, N_NODES, OUT_DIM);
    }

    // --- global mean pool ---
    fill_f32<<<cdiv((long)N_GRAPHS * 128 + N_GRAPHS, B), B, 0, stream>>>(
        sums, 0.0f, (long)N_GRAPHS * 128 + N_GRAPHS);   // sums + counts contiguous
    pool_accum<<<cdiv((long)N_NODES * (OUT_DIM / 4), B), B, 0, stream>>>(
        batch, ACT, sums, cnts, N_NODES, OUT_DIM);
    pool_div<<<cdiv((long)N_GRAPHS * OUT_DIM, B), B, 0, stream>>>(
        sums, cnts, out, N_GRAPHS, OUT_DIM);
}